// DruseScorePKi_68710886802024
// MI455X (gfx1250) — compile-verified
//
#include <hip/hip_runtime.h>
#include <math.h>

// ---------------- problem constants ----------------
#define B_        8
#define P_        350
#define L_        55
#define ATOM_     18
#define HID_      128
#define HEADS_    4
#define HEAD_DIM_ 32
#define RBF_DIM_  50
#define MAXN_     32
#define NPROT     (B_*P_)        // 2800
#define NLIG      (B_*L_)        // 440
#define EPROT     (B_*P_*MAXN_)  // 89600
#define ELIG      (B_*L_*MAXN_)  // 14080
#define FARV      1000000.0f
#define GAMMA_    10.0f

typedef float v2f __attribute__((ext_vector_type(2)));
typedef float v8f __attribute__((ext_vector_type(8)));

__device__ __forceinline__ float silu_f(float x) { return x / (1.f + expf(-x)); }
__device__ __forceinline__ float sigmoid_f(float x) { return 1.f / (1.f + expf(-x)); }

// ---------------- WMMA f32 16x16x4 ----------------
__device__ __forceinline__ v8f wmma4(v2f a, v2f b, v8f c) {
  // D = A(16x4 f32) * B(4x16 f32) + C(16x16 f32)
  return __builtin_amdgcn_wmma_f32_16x16x4_f32(false, a, false, b, (short)0, c,
                                               false, false);
}

// Wave-level GEMM, N fixed at 128: C_tile(16 x 128) = As(16 x KP, LDS) * Wpk.
// Wpk is the pair-packed, zero-padded weight: Wpk[(k>>1)*256 + col*2 + (k&1)] = W[k][col],
// so each lane fetches its (k, k+1) pair with a single 8-byte load.
// 4 waves per block; wave w computes output column tiles w and w+4 concurrently
// (two accumulators sharing one A fragment -> half the DS traffic, 2 indep. WMMA chains).
// act: 0 = none, 1 = SiLU.
__device__ __forceinline__ void gemm_16x128(
    const float* __restrict__ As, int KP,
    const float* __restrict__ Wpk, const float* __restrict__ bias,
    float* __restrict__ C, int Cstride, int rowBase, int rowLimit,
    int act, int lane, int wave)
{
  const int r  = lane & 15;          // A row / output column-within-tile
  const int kh = (lane >> 4) << 1;   // 0 or 2 (K pair select)
  const int rb = (lane >> 4) << 3;   // 0 or 8 (output row half)
  const float* Arow  = As + r * KP + kh;
  const int col0 = (wave << 4) + r;
  const int col1 = col0 + 64;
  const float* wbase = Wpk + (size_t)(kh >> 1) * 256;   // pair row for this lane half
  v8f acc0 = {0.f,0.f,0.f,0.f,0.f,0.f,0.f,0.f};
  v8f acc1 = {0.f,0.f,0.f,0.f,0.f,0.f,0.f,0.f};
#pragma unroll 4
  for (int k = 0; k < KP; k += 4) {
    v2f a; a.x = Arow[k];
           a.y = Arow[k + 1];
    const float* wk = wbase + (size_t)(k >> 1) * 256;
    const v2f b0 = *(const v2f*)(wk + col0 * 2);
    const v2f b1 = *(const v2f*)(wk + col1 * 2);
    acc0 = wmma4(a, b0, acc0);
    acc1 = wmma4(a, b1, acc1);
  }
  const float bv0 = bias ? bias[col0] : 0.f;
  const float bv1 = bias ? bias[col1] : 0.f;
#pragma unroll
  for (int g = 0; g < 8; ++g) {
    const int m = rowBase + rb + g;
    if (m < rowLimit) {
      float v0 = acc0[g] + bv0;
      float v1 = acc1[g] + bv1;
      if (act) { v0 = v0 / (1.f + expf(-v0)); v1 = v1 / (1.f + expf(-v1)); }
      C[(size_t)m * Cstride + col0] = v0;
      C[(size_t)m * Cstride + col1] = v1;
    }
  }
}

// ---------------- weight pack kernel: pair-interleave + zero-pad ----------------
// WP[(k>>1)*N*2 + n*2 + (k&1)] = (k < K) ? W[k*N + n] : 0   for k in [0, KP)
__global__ void pack_w_kernel(const float* __restrict__ W, float* __restrict__ WP,
                              int K, int KP, int N) {
  int i = blockIdx.x * blockDim.x + threadIdx.x;
  if (i >= KP * N) return;
  const int k = i / N, n = i - k * N;
  const float v = (k < K) ? W[(size_t)k * N + n] : 0.f;
  WP[((size_t)(k >> 1) * N + n) * 2 + (k & 1)] = v;
}

// ---------------- generic GEMM kernel: C(Mx128) = A(MxK) @ W + bias ----------------
// blockDim = 128 (4 waves). grid.x = ceil(M/16). K <= 256. Wpk pre-packed.
__global__ void gemm_kernel(const float* __restrict__ A, const float* __restrict__ Wpk,
                            const float* __restrict__ bias, float* __restrict__ C,
                            int M, int K, int act)
{
  __shared__ float As[16 * 256];
  const int tid = threadIdx.x, lane = tid & 31, wave = tid >> 5;
  const int m0 = blockIdx.x * 16;
  const int KP = (K + 3) & ~3;
  for (int idx = tid; idx < 16 * KP; idx += 128) {
    const int rr = idx / KP, cc = idx - rr * KP;
    const int m = m0 + rr;
    As[idx] = (m < M && cc < K) ? A[(size_t)m * K + cc] : 0.f;
  }
  __syncthreads();
  gemm_16x128(As, KP, Wpk, bias, C, HID_, m0, M, act, lane, wave);
}

// ---------------- elementwise helpers ----------------
__global__ void fill_kernel(float* __restrict__ p, float v, int n) {
  int i = blockIdx.x * blockDim.x + threadIdx.x;
  if (i < n) p[i] = v;
}
__global__ void copy_kernel(float* __restrict__ dst, const float* __restrict__ src, int n) {
  int i = blockIdx.x * blockDim.x + threadIdx.x;
  if (i < n) dst[i] = src[i];
}
__global__ void mask_pos_kernel(const float* __restrict__ pos, const float* __restrict__ mask,
                                float* __restrict__ out, int nNodes) {
  int i = blockIdx.x * blockDim.x + threadIdx.x;
  if (i < nNodes * 3) out[i] = (mask[i / 3] > 0.f) ? pos[i] : FARV;
}
__global__ void concat2_kernel(const float* __restrict__ a, const float* __restrict__ b,
                               float* __restrict__ out, int M) {
  int i = blockIdx.x * blockDim.x + threadIdx.x;
  if (i >= M * 2 * HID_) return;
  int m = i >> 8, c = i & 255;
  out[i] = (c < HID_) ? a[m * HID_ + c] : b[m * HID_ + (c - HID_)];
}

// ---------------- LayerNorm(h + hn) * g + b  (block per row, 128 threads) ----------------
__global__ void ln_res_kernel(const float* __restrict__ h, const float* __restrict__ hn,
                              const float* __restrict__ g, const float* __restrict__ bb,
                              float* __restrict__ out, int M)
{
  const int m = blockIdx.x, c = threadIdx.x;
  __shared__ float red[HID_];
  float x = h[(size_t)m * HID_ + c] + hn[(size_t)m * HID_ + c];
  red[c] = x; __syncthreads();
  for (int s = 64; s > 0; s >>= 1) { if (c < s) red[c] += red[c + s]; __syncthreads(); }
  const float mean = red[0] * (1.f / HID_);
  __syncthreads();
  const float d = x - mean;
  red[c] = d * d; __syncthreads();
  for (int s = 64; s > 0; s >>= 1) { if (c < s) red[c] += red[c + s]; __syncthreads(); }
  const float var = red[0] * (1.f / HID_);
  out[(size_t)m * HID_ + c] = d * rsqrtf(var + 1e-5f) * g[c] + bb[c];
}

// ---------------- EGNN edge kernel (16 edges per block, 128 threads) ----------------
// e1w/e2w/c1w are the pre-packed copies.
__global__ void egnn_edge_kernel(const float* __restrict__ h, const float* __restrict__ pos,
    const int* __restrict__ erow, const int* __restrict__ ecol,
    const float* __restrict__ e1w, const float* __restrict__ e1b,
    const float* __restrict__ e2w, const float* __restrict__ e2b,
    const float* __restrict__ c1w, const float* __restrict__ c1b,
    const float* __restrict__ c2w,
    float* __restrict__ agg, float* __restrict__ posOut)
{
  __shared__ float Ein[16 * 260];   // [hrow(128) | hcol(128) | dist | pad3]
  __shared__ float Hid[16 * 128];
  __shared__ float Msg[16 * 128];
  __shared__ float Dif[16 * 4];
  __shared__ float Cw[16];
  __shared__ int   Rows[16];
  __shared__ int   Cols[16];
  const int tid = threadIdx.x, lane = tid & 31, wave = tid >> 5;
  const int e0 = blockIdx.x * 16;

  if (tid < 16) {
    const int e = e0 + tid;
    const int r = erow[e], c = ecol[e];
    Rows[tid] = r; Cols[tid] = c;
    const float dx = pos[r * 3 + 0] - pos[c * 3 + 0];
    const float dy = pos[r * 3 + 1] - pos[c * 3 + 1];
    const float dz = pos[r * 3 + 2] - pos[c * 3 + 2];
    float d = sqrtf(dx * dx + dy * dy + dz * dz + 1e-12f);
    d = fmaxf(d, 1e-6f);
    Dif[tid * 4 + 0] = dx; Dif[tid * 4 + 1] = dy; Dif[tid * 4 + 2] = dz; Dif[tid * 4 + 3] = d;
    Ein[tid * 260 + 256] = d;
    Ein[tid * 260 + 257] = 0.f; Ein[tid * 260 + 258] = 0.f; Ein[tid * 260 + 259] = 0.f;
  }
  __syncthreads();
  for (int idx = tid; idx < 16 * HID_; idx += 128) {
    const int i = idx >> 7, ch = idx & 127;
    Ein[i * 260 + ch]        = h[(size_t)Rows[i] * HID_ + ch];
    Ein[i * 260 + HID_ + ch] = h[(size_t)Cols[i] * HID_ + ch];
  }
  __syncthreads();
  // msg = silu(silu(e_in @ e1 + b1) @ e2 + b2)
  gemm_16x128(Ein, 260, e1w, e1b, Hid, HID_, 0, 16, 1, lane, wave);
  __syncthreads();
  gemm_16x128(Hid, 128, e2w, e2b, Msg, HID_, 0, 16, 1, lane, wave);
  __syncthreads();
  // cw = clip(silu(msg @ c1 + b) @ c2, -1, 1)
  gemm_16x128(Msg, 128, c1w, c1b, Hid, HID_, 0, 16, 1, lane, wave);
  __syncthreads();
  if (tid < 16) {
    float s = 0.f;
    for (int k = 0; k < HID_; ++k) s += Hid[tid * HID_ + k] * c2w[k];
    Cw[tid] = fminf(1.f, fmaxf(-1.f, s));
  }
  __syncthreads();
  // agg[row] += msg ; posOut[row] += diff/dist * cw
  for (int idx = tid; idx < 16 * HID_; idx += 128) {
    const int i = idx >> 7, ch = idx & 127;
    unsafeAtomicAdd(&agg[(size_t)Rows[i] * HID_ + ch], Msg[i * HID_ + ch]);
  }
  if (tid < 48) {
    const int i = tid / 3, c = tid - i * 3;
    const float upd = Dif[i * 4 + c] / Dif[i * 4 + 3] * Cw[i];
    unsafeAtomicAdd(&posOut[(size_t)Rows[i] * 3 + c], upd);
  }
}

// ---------------- cross attention (block per (b,l), 128 threads) ----------------
__global__ void cross_attn_kernel(const float* __restrict__ Q, const float* __restrict__ Kb,
    const float* __restrict__ Vb, const float* __restrict__ ligPe,
    const float* __restrict__ protPe, const float* __restrict__ protMask,
    const float* __restrict__ rbfW, const float* __restrict__ rbfB,
    float* __restrict__ attOut, float* __restrict__ attnWOut)
{
  const int bl = blockIdx.x;                  // b*L + l
  const int b = bl / L_;
  const int tid = threadIdx.x;
  __shared__ float Qs[HID_];
  __shared__ float sc[P_ * HEADS_];
  Qs[tid] = Q[(size_t)bl * HID_ + tid];
  __syncthreads();
  const float lx = ligPe[bl * 3 + 0], ly = ligPe[bl * 3 + 1], lz = ligPe[bl * 3 + 2];
  const float scale = 1.f / sqrtf((float)HEAD_DIM_);
  for (int p = tid; p < P_; p += 128) {
    const int gp = b * P_ + p;
    const float dx = lx - protPe[gp * 3 + 0];
    const float dy = ly - protPe[gp * 3 + 1];
    const float dz = lz - protPe[gp * 3 + 2];
    const float dist = sqrtf(dx * dx + dy * dy + dz * dz + 1e-12f);
    float bias0 = rbfB[0], bias1 = rbfB[1], bias2 = rbfB[2], bias3 = rbfB[3];
    for (int j = 0; j < RBF_DIM_; ++j) {
      const float t = dist - (10.f * (float)j / 49.f);
      const float rv = expf(-GAMMA_ * t * t);
      bias0 += rv * rbfW[j * 4 + 0]; bias1 += rv * rbfW[j * 4 + 1];
      bias2 += rv * rbfW[j * 4 + 2]; bias3 += rv * rbfW[j * 4 + 3];
    }
    const bool valid = protMask[gp] > 0.f;
    float biasH[4] = {bias0, bias1, bias2, bias3};
#pragma unroll
    for (int hh = 0; hh < HEADS_; ++hh) {
      float s = 0.f;
      for (int d = 0; d < HEAD_DIM_; ++d)
        s += Qs[hh * HEAD_DIM_ + d] * Kb[(size_t)gp * HID_ + hh * HEAD_DIM_ + d];
      s = s * scale + biasH[hh];
      sc[p * HEADS_ + hh] = valid ? s : -1e9f;
    }
  }
  __syncthreads();
  // per-head softmax over P: wave w handles head w
  const int lane = tid & 31, wave = tid >> 5;
  float mx = -1e30f;
  for (int p = lane; p < P_; p += 32) mx = fmaxf(mx, sc[p * HEADS_ + wave]);
  for (int o = 16; o; o >>= 1) mx = fmaxf(mx, __shfl_xor(mx, o, 32));
  float sum = 0.f;
  for (int p = lane; p < P_; p += 32) {
    const float e = expf(sc[p * HEADS_ + wave] - mx);
    sc[p * HEADS_ + wave] = e;
    sum += e;
  }
  for (int o = 16; o; o >>= 1) sum += __shfl_xor(sum, o, 32);
  const float inv = 1.f / sum;
  for (int p = lane; p < P_; p += 32) sc[p * HEADS_ + wave] *= inv;
  __syncthreads();
  for (int p = tid; p < P_; p += 128)
    attnWOut[(size_t)bl * P_ + p] = 0.25f * (sc[p * 4] + sc[p * 4 + 1] + sc[p * 4 + 2] + sc[p * 4 + 3]);
  // out[c] = sum_p attn[p, c/32] * V[p, c]
  const int hd = tid >> 5;
  float acc = 0.f;
  for (int p = 0; p < P_; ++p)
    acc += sc[p * HEADS_ + hd] * Vb[((size_t)b * P_ + p) * HID_ + tid];
  attOut[(size_t)bl * HID_ + tid] = acc;
}

// ---------------- pooling + affinity/confidence heads (block per batch) ----------------
__global__ void heads_kernel(const float* __restrict__ ligAtt, const float* __restrict__ ligMask,
    const float* __restrict__ a1w, const float* __restrict__ a1b,
    const float* __restrict__ a2w, const float* __restrict__ a2b,
    const float* __restrict__ c1w, const float* __restrict__ c1b,
    const float* __restrict__ c2w, const float* __restrict__ c2b,
    float* __restrict__ out)
{
  const int b = blockIdx.x, c = threadIdx.x;
  __shared__ float repr[HID_], h1[HID_], h2[64], red[HID_], scal[1];
  float s = 0.f, msum = 0.f;
  for (int l = 0; l < L_; ++l) {
    const float mk = ligMask[b * L_ + l];
    msum += mk;
    s += ligAtt[((size_t)b * L_ + l) * HID_ + c] * mk;
  }
  repr[c] = s / fmaxf(msum, 1.f);
  __syncthreads();
  float acc = a1b[c];
  for (int k = 0; k < HID_; ++k) acc += repr[k] * a1w[k * HID_ + c];
  h1[c] = silu_f(acc);
  if (c < 64) {
    float a = c1b[c];
    for (int k = 0; k < HID_; ++k) a += repr[k] * c1w[k * 64 + c];
    h2[c] = silu_f(a);
  }
  __syncthreads();
  red[c] = h1[c] * a2w[c]; __syncthreads();
  for (int st = 64; st > 0; st >>= 1) { if (c < st) red[c] += red[c + st]; __syncthreads(); }
  if (c == 0) scal[0] = red[0] + a2b[0];
  __syncthreads();
  red[c] = (c < 64) ? h2[c] * c2w[c] : 0.f; __syncthreads();
  for (int st = 64; st > 0; st >>= 1) { if (c < st) red[c] += red[c + st]; __syncthreads(); }
  if (c == 0) {
    const float pkd = scal[0];
    const float conf = sigmoid_f(red[0] + c2b[0]);
    out[b] = pkd; out[8 + b] = conf; out[16 + b] = pkd * conf;
  }
}

// ---------------- pair / interaction MLP (16 pairs per block, WMMA) ----------------
// i1w is the pre-packed 308-row copy.
__global__ void pair_kernel(const float* __restrict__ ligAtt, const float* __restrict__ protH,
    const float* __restrict__ ligPe, const float* __restrict__ protPe,
    const float* __restrict__ i1w, const float* __restrict__ i1b,
    const float* __restrict__ i2w, const float* __restrict__ i2b,
    float* __restrict__ interOut)
{
  __shared__ float Ein[16 * 308];   // [ligAtt(128) | protH(128) | rbf(50) | pad2]
  __shared__ float Hid[16 * 128];
  const int tid = threadIdx.x, lane = tid & 31, wave = tid >> 5;
  const int t0 = blockIdx.x * 16;
  for (int idx = tid; idx < 16 * HID_; idx += 128) {
    const int i = idx >> 7, ch = idx & 127;
    const int pi = t0 + i;
    const int b = pi / (L_ * P_);
    const int rem = pi - b * (L_ * P_);
    const int l = rem / P_, p = rem - l * P_;
    Ein[i * 308 + ch]        = ligAtt[((size_t)b * L_ + l) * HID_ + ch];
    Ein[i * 308 + HID_ + ch] = protH[((size_t)b * P_ + p) * HID_ + ch];
  }
  if (tid < 16) {
    const int pi = t0 + tid;
    const int b = pi / (L_ * P_);
    const int rem = pi - b * (L_ * P_);
    const int l = rem / P_, p = rem - l * P_;
    const int gl = b * L_ + l, gp = b * P_ + p;
    const float dx = ligPe[gl * 3 + 0] - protPe[gp * 3 + 0];
    const float dy = ligPe[gl * 3 + 1] - protPe[gp * 3 + 1];
    const float dz = ligPe[gl * 3 + 2] - protPe[gp * 3 + 2];
    const float dist = sqrtf(dx * dx + dy * dy + dz * dz + 1e-12f);
    for (int j = 0; j < RBF_DIM_; ++j) {
      const float t = dist - (10.f * (float)j / 49.f);
      Ein[tid * 308 + 256 + j] = expf(-GAMMA_ * t * t);
    }
    Ein[tid * 308 + 306] = 0.f; Ein[tid * 308 + 307] = 0.f;
  }
  __syncthreads();
  gemm_16x128(Ein, 308, i1w, i1b, Hid, HID_, 0, 16, 1, lane, wave);
  __syncthreads();
  for (int idx = tid; idx < 16 * 5; idx += 128) {
    const int i = idx / 5, j = idx - i * 5;
    float a = i2b[j];
    for (int k = 0; k < HID_; ++k) a += Hid[i * HID_ + k] * i2w[k * 5 + j];
    interOut[(size_t)(t0 + i) * 5 + j] = sigmoid_f(a);
  }
}

// ---------------- host orchestration ----------------
extern "C" void kernel_launch(void* const* d_in, const int* in_sizes, int n_in,
                              void* d_out, int out_size, void* d_ws, size_t ws_size,
                              hipStream_t stream)
{
  (void)in_sizes; (void)n_in; (void)out_size; (void)ws_size;
  const float* prot_x    = (const float*)d_in[0];
  const float* prot_pos  = (const float*)d_in[1];
  const float* prot_mask = (const float*)d_in[2];
  const float* lig_x     = (const float*)d_in[3];
  const float* lig_pos   = (const float*)d_in[4];
  const float* lig_mask  = (const float*)d_in[5];
  const int*   prot_edges= (const int*)d_in[6];
  const int*   lig_edges = (const int*)d_in[7];

  // params flattened as JAX pytree leaves (sorted dict keys), starting at d_in[8]
  enum {
    AFF1_B = 8, AFF1_W, AFF2_B, AFF2_W, CA_LN_B, CA_LN_G,
    CONF1_B, CONF1_W, CONF2_B, CONF2_W,
    INT1_B, INT1_W, INT2_B, INT2_W,
    KK_B, KK_W, LIG_IN_B, LIG_IN_W, LIG_L0 = 26,
    O_B = 78, O_W, PROT_IN_B = 80, PROT_IN_W, PROT_L0 = 82,
    Q_B = 134, Q_W, RBF_B = 136, RBF_W, V_B = 138, V_W
  };
  // per-layer leaf offsets (sorted: c1,c2,e1,e2,ln_b,ln_g,n1,n2 ; each lin: b,w ; c2 has w only)
  enum { LY_C1B = 0, LY_C1W, LY_C2W, LY_E1B, LY_E1W, LY_E2B, LY_E2W,
         LY_LNB, LY_LNG, LY_N1B, LY_N1W, LY_N2B, LY_N2W, LY_STRIDE };
  auto F = [&](int i) { return (const float*)d_in[i]; };
  auto cdiv = [](int a, int b) { return (a + b - 1) / b; };

  // workspace carve-out
  float* ws = (float*)d_ws;
  size_t off = 0;
  auto alloc = [&](size_t n) { float* p = ws + off; off += n; return p; };
  float* ph    = alloc((size_t)NPROT * HID_);
  float* ptmp  = alloc((size_t)NPROT * HID_);
  float* pagg  = alloc((size_t)NPROT * HID_);
  float* pcat  = alloc((size_t)NPROT * 2 * HID_);
  float* pposA = alloc((size_t)NPROT * 3);
  float* pposB = alloc((size_t)NPROT * 3);
  float* lh    = alloc((size_t)NLIG * HID_);
  float* ltmp  = alloc((size_t)NLIG * HID_);
  float* lagg  = alloc((size_t)NLIG * HID_);
  float* lcat  = alloc((size_t)NLIG * 2 * HID_);
  float* lposA = alloc((size_t)NLIG * 3);
  float* lposB = alloc((size_t)NLIG * 3);
  float* Qb    = alloc((size_t)NLIG * HID_);
  float* Kb    = alloc((size_t)NPROT * HID_);
  float* Vb    = alloc((size_t)NPROT * HID_);
  float* attO  = alloc((size_t)NLIG * HID_);
  float* ligAtt= alloc((size_t)NLIG * HID_);
  // pair-packed (and zero-padded) weight copies for all WMMA GEMMs
  float* in_pk[2];   for (int i = 0; i < 2; ++i) in_pk[i] = alloc((size_t)20 * HID_);
  float* int1_pk = alloc((size_t)308 * HID_);
  float* q_pk = alloc((size_t)128 * HID_);
  float* k_pk = alloc((size_t)128 * HID_);
  float* v_pk = alloc((size_t)128 * HID_);
  float* o_pk = alloc((size_t)128 * HID_);
  float* e1_pk[8]; float* e2_pk[8]; float* c1_pk[8]; float* n1_pk[8]; float* n2_pk[8];
  for (int i = 0; i < 8; ++i) {
    e1_pk[i] = alloc((size_t)260 * HID_);
    e2_pk[i] = alloc((size_t)128 * HID_);
    c1_pk[i] = alloc((size_t)128 * HID_);
    n1_pk[i] = alloc((size_t)256 * HID_);
    n2_pk[i] = alloc((size_t)128 * HID_);
  }

  auto pack = [&](const float* src, float* dst, int K, int KP) {
    pack_w_kernel<<<cdiv(KP * HID_, 256), 256, 0, stream>>>(src, dst, K, KP, HID_);
  };
  // ---- build packed weights (once per launch, tiny) ----
  pack(F(PROT_IN_W), in_pk[0], ATOM_, 20);
  pack(F(LIG_IN_W),  in_pk[1], ATOM_, 20);
  pack(F(INT1_W), int1_pk, 306, 308);
  pack(F(Q_W), q_pk, 128, 128);
  pack(F(KK_W), k_pk, 128, 128);
  pack(F(V_W), v_pk, 128, 128);
  pack(F(O_W), o_pk, 128, 128);
  for (int e = 0; e < 2; ++e) {
    const int lay0 = e ? LIG_L0 : PROT_L0;
    for (int li = 0; li < 4; ++li) {
      const int base = lay0 + li * LY_STRIDE;
      const int idx = e * 4 + li;
      pack(F(base + LY_E1W), e1_pk[idx], 257, 260);
      pack(F(base + LY_E2W), e2_pk[idx], 128, 128);
      pack(F(base + LY_C1W), c1_pk[idx], 128, 128);
      pack(F(base + LY_N1W), n1_pk[idx], 256, 256);
      pack(F(base + LY_N2W), n2_pk[idx], 128, 128);
    }
  }

  auto run_encoder = [&](const float* x, const float* pos, const float* mask,
                         const int* edges, int nNodes, int nEdges,
                         int inB, const float* in_wp, int lay0, int pkBase,
                         float* h, float* tmp, float* agg, float* cat,
                         float* posA, float* posB) -> float* {
    const int mt = cdiv(nNodes, 16);
    mask_pos_kernel<<<cdiv(nNodes * 3, 256), 256, 0, stream>>>(pos, mask, posA, nNodes);
    gemm_kernel<<<mt, 128, 0, stream>>>(x, in_wp, F(inB), h, nNodes, ATOM_, 0);
    float* cur = posA; float* nxt = posB;
    for (int li = 0; li < 4; ++li) {
      const int base = lay0 + li * LY_STRIDE;
      const int idx = pkBase + li;
      fill_kernel<<<cdiv(nNodes * HID_, 256), 256, 0, stream>>>(agg, 0.f, nNodes * HID_);
      copy_kernel<<<cdiv(nNodes * 3, 256), 256, 0, stream>>>(nxt, cur, nNodes * 3);
      egnn_edge_kernel<<<nEdges / 16, 128, 0, stream>>>(h, cur, edges, edges + nEdges,
          e1_pk[idx], F(base + LY_E1B), e2_pk[idx], F(base + LY_E2B),
          c1_pk[idx], F(base + LY_C1B), F(base + LY_C2W), agg, nxt);
      concat2_kernel<<<cdiv(nNodes * 2 * HID_, 256), 256, 0, stream>>>(h, agg, cat, nNodes);
      gemm_kernel<<<mt, 128, 0, stream>>>(cat, n1_pk[idx], F(base + LY_N1B), tmp,
                                          nNodes, 2 * HID_, 1);
      gemm_kernel<<<mt, 128, 0, stream>>>(tmp, n2_pk[idx], F(base + LY_N2B), agg,
                                          nNodes, HID_, 0);
      ln_res_kernel<<<nNodes, 128, 0, stream>>>(h, agg, F(base + LY_LNG), F(base + LY_LNB), h, nNodes);
      float* t = cur; cur = nxt; nxt = t;
    }
    return cur;
  };

  float* prot_pe = run_encoder(prot_x, prot_pos, prot_mask, prot_edges, NPROT, EPROT,
                               PROT_IN_B, in_pk[0], PROT_L0, 0,
                               ph, ptmp, pagg, pcat, pposA, pposB);
  float* lig_pe  = run_encoder(lig_x, lig_pos, lig_mask, lig_edges, NLIG, ELIG,
                               LIG_IN_B, in_pk[1], LIG_L0, 4,
                               lh, ltmp, lagg, lcat, lposA, lposB);

  // cross attention
  gemm_kernel<<<cdiv(NLIG, 16), 128, 0, stream>>>(lh, q_pk, F(Q_B), Qb, NLIG, HID_, 0);
  gemm_kernel<<<NPROT / 16, 128, 0, stream>>>(ph, k_pk, F(KK_B), Kb, NPROT, HID_, 0);
  gemm_kernel<<<NPROT / 16, 128, 0, stream>>>(ph, v_pk, F(V_B), Vb, NPROT, HID_, 0);

  float* out       = (float*)d_out;
  float* out_inter = out + 24;
  float* out_attnw = out + 24 + (size_t)B_ * L_ * P_ * 5;

  cross_attn_kernel<<<NLIG, 128, 0, stream>>>(Qb, Kb, Vb, lig_pe, prot_pe, prot_mask,
                                              F(RBF_W), F(RBF_B), attO, out_attnw);
  gemm_kernel<<<cdiv(NLIG, 16), 128, 0, stream>>>(attO, o_pk, F(O_B), ltmp, NLIG, HID_, 0);
  ln_res_kernel<<<NLIG, 128, 0, stream>>>(lh, ltmp, F(CA_LN_G), F(CA_LN_B), ligAtt, NLIG);

  heads_kernel<<<B_, 128, 0, stream>>>(ligAtt, lig_mask,
      F(AFF1_W), F(AFF1_B), F(AFF2_W), F(AFF2_B),
      F(CONF1_W), F(CONF1_B), F(CONF2_W), F(CONF2_B), out);

  pair_kernel<<<(B_ * L_ * P_) / 16, 128, 0, stream>>>(ligAtt, ph, lig_pe, prot_pe,
      int1_pk, F(INT1_B), F(INT2_W), F(INT2_B), out_inter);
}